// RetinaUNetLayer_26285199851828
// MI455X (gfx1250) — compile-verified
//
#include <hip/hip_runtime.h>
#include <stdint.h>

// RetinaUNet post-process for gfx1250 (MI455X, wave32):
//   decode+clip -> bitonic argsort(desc) -> WMMA-assisted IoU bitmask ->
//   single-wave register sweep -> scatter to original order.
//
// IoU test rewritten division-free so the additive part maps to WMMA:
//   inter/(union+eps) > thr  <=>  inter*(1+thr) > thr*(ai+aj) + thr*eps
// RHS computed per 16x16 tile by V_WMMA_F32_16X16X4_F32 (A=[thr*ai, thr],
// B=[1; aj], C=thr*eps).

#define N_BOX   8192
#define NWORDS  (N_BOX / 32)      // 256 mask words per row
#define WIN     512.0f
#define THR     0.5f
#define EPS_F   1e-6f

typedef __attribute__((ext_vector_type(2))) float v2f;
typedef __attribute__((ext_vector_type(8))) float v8f;

// ---------------- 1) decode + clip ----------------
__global__ __launch_bounds__(256) void decode_kernel(const float4* __restrict__ anchors,
                                                     const float4* __restrict__ deltas,
                                                     float4* __restrict__ boxes) {
    int i = blockIdx.x * blockDim.x + threadIdx.x;
    if (i >= N_BOX) return;
    float4 a = anchors[i];   // y1,x1,y2,x2
    float4 d = deltas[i];    // dy,dx,dh,dw
    float h  = a.z - a.x, w = a.w - a.y;
    float cy = a.x + 0.5f * h, cx = a.y + 0.5f * w;
    float pcy = d.x * h + cy, pcx = d.y * w + cx;
    float ph  = expf(d.z) * h, pw = expf(d.w) * w;
    float4 b;
    b.x = fminf(fmaxf(pcy - 0.5f * ph, 0.f), WIN);
    b.y = fminf(fmaxf(pcx - 0.5f * pw, 0.f), WIN);
    b.z = fminf(fmaxf(pcy + 0.5f * ph, 0.f), WIN);
    b.w = fminf(fmaxf(pcx + 0.5f * pw, 0.f), WIN);
    boxes[i] = b;
}

// ---------------- 2) stable descending argsort (bitonic, dyn-LDS) --------
__global__ __launch_bounds__(1024) void sort_kernel(const float* __restrict__ scores,
                                                    const float4* __restrict__ boxes,
                                                    uint32_t* __restrict__ order,
                                                    float4* __restrict__ bsort,
                                                    float* __restrict__ areas) {
    extern __shared__ unsigned long long keys[];   // 8192 * 8B = 64 KB
    const int tid = threadIdx.x;
    for (int i = tid; i < N_BOX; i += 1024) {
        unsigned sb = __float_as_uint(scores[i]);  // scores in [0,1): bits monotonic
        keys[i] = ((unsigned long long)sb << 32) | (unsigned)(~(unsigned)i);
    }
    for (unsigned size = 2; size <= N_BOX; size <<= 1) {
        for (unsigned str = size >> 1; str > 0; str >>= 1) {
            __syncthreads();
            for (unsigned t = tid; t < N_BOX / 2; t += 1024) {
                unsigned lo = 2u * t - (t & (str - 1u));
                unsigned hi = lo + str;
                bool desc = ((lo & size) == 0u);
                unsigned long long ka = keys[lo], kb = keys[hi];
                if ((ka < kb) == desc) { keys[lo] = kb; keys[hi] = ka; }
            }
        }
    }
    __syncthreads();
    for (int i = tid; i < N_BOX; i += 1024) {
        unsigned idx = ~(unsigned)(keys[i] & 0xFFFFFFFFull);
        order[i] = idx;
        float4 b = boxes[idx];
        bsort[i] = b;
        areas[i] = fmaxf(b.z - b.x, 0.f) * fmaxf(b.w - b.y, 0.f);
    }
}

// ---------------- 3) suppression bitmask: one wave per 16x32 tile --------
__global__ __launch_bounds__(32) void mask_kernel(const float4* __restrict__ bsort,
                                                  const float* __restrict__ areas,
                                                  uint32_t* __restrict__ mask) {
    const int lane = threadIdx.x;
    const int i0 = blockIdx.y * 16;      // row tile
    const int j0 = blockIdx.x * 32;      // col word
    __shared__ float4 rb[16];
    __shared__ float  ra[16];
    if (lane < 16) {
        rb[lane] = bsort[i0 + lane];
        ra[lane] = areas[i0 + lane];
    }
    __syncthreads();

    const bool lo = lane < 16;
    const int  ls = lane & 15;

    // A (16x4 f32): lanes 0-15: K0 = thr*area_row[M=lane], K1 = thr ; K2,K3 = 0
    v2f A; A.x = lo ? THR * ra[lane] : 0.f; A.y = lo ? THR : 0.f;
    float ca0 = areas[j0 + ls];
    float ca1 = areas[j0 + 16 + ls];
    // B (4x16 f32): K0 row = ones, K1 row = area_col[N=lane] ; K2,K3 = 0
    v2f B0; B0.x = lo ? 1.f : 0.f; B0.y = lo ? ca0 : 0.f;
    v2f B1; B1.x = lo ? 1.f : 0.f; B1.y = lo ? ca1 : 0.f;
    v8f C;
#pragma unroll
    for (int k = 0; k < 8; ++k) C[k] = THR * EPS_F;

    // D[i][j] = thr*area_i + thr*area_j + thr*eps
    v8f D0 = __builtin_amdgcn_wmma_f32_16x16x4_f32(false, A, false, B0, (short)0, C, false, false);
    v8f D1 = __builtin_amdgcn_wmma_f32_16x16x4_f32(false, A, false, B1, (short)0, C, false, false);

    float4 c0 = bsort[j0 + ls];
    float4 c1 = bsort[j0 + 16 + ls];

#pragma unroll
    for (int r = 0; r < 8; ++r) {
        int M = lo ? r : (8 + r);        // C/D layout: lanes<16 -> M=r, lanes>=16 -> M=8+r
        float4 bm = rb[M];
        float ih0 = fminf(bm.z, c0.z) - fmaxf(bm.x, c0.x);
        float iw0 = fminf(bm.w, c0.w) - fmaxf(bm.y, c0.y);
        float inter0 = fmaxf(ih0, 0.f) * fmaxf(iw0, 0.f);
        float ih1 = fminf(bm.z, c1.z) - fmaxf(bm.x, c1.x);
        float iw1 = fminf(bm.w, c1.w) - fmaxf(bm.y, c1.y);
        float inter1 = fmaxf(ih1, 0.f) * fmaxf(iw1, 0.f);
        int rowg = i0 + M;
        bool p0 = (inter0 * (1.f + THR) > D0[r]) && ((j0 + ls)      != rowg); // kill diagonal
        bool p1 = (inter1 * (1.f + THR) > D1[r]) && ((j0 + 16 + ls) != rowg);
        unsigned bal0 = (unsigned)__ballot(p0);  // bits 0-15: row r ; bits 16-31: row 8+r
        unsigned bal1 = (unsigned)__ballot(p1);
        if (lane == 0) {
            mask[(size_t)(i0 + r)     * NWORDS + blockIdx.x] =
                (bal0 & 0xFFFFu) | ((bal1 & 0xFFFFu) << 16);
            mask[(size_t)(i0 + 8 + r) * NWORDS + blockIdx.x] =
                (bal0 >> 16) | (bal1 & 0xFFFF0000u);
        }
    }
}

// ---------------- 4) sequential sweep: one wave, state in VGPRs ----------
__global__ __launch_bounds__(32) void sweep_kernel(const uint32_t* __restrict__ mask,
                                                   uint32_t* __restrict__ keepWords) {
    const int lane = threadIdx.x;
    uint32_t remv[8] = {0, 0, 0, 0, 0, 0, 0, 0};   // word (slot*32+lane)
#pragma unroll
    for (int slot = 0; slot < 8; ++slot) {
        for (int w32 = 0; w32 < 32; ++w32) {
            const int w = slot * 32 + w32;
            uint32_t cw = __shfl(remv[slot], w32, 32);
            for (int bit = 0; bit < 32; ++bit) {
                if (!((cw >> bit) & 1u)) {
                    const int i = (w << 5) | bit;            // kept box: OR its row
                    const uint32_t* row = mask + (size_t)i * NWORDS;
#pragma unroll
                    for (int s = 0; s < 8; ++s) remv[s] |= row[s * 32 + lane];
                    cw = __shfl(remv[slot], w32, 32);        // refresh current word
                }
            }
        }
    }
    // A kept box's bit can never be set afterwards (greedy + symmetry), so:
#pragma unroll
    for (int slot = 0; slot < 8; ++slot)
        keepWords[slot * 32 + lane] = ~remv[slot];
}

// ---------------- 5) scatter back to original order ----------------------
__global__ __launch_bounds__(256) void finalize_kernel(const uint32_t* __restrict__ order,
                                                       const uint32_t* __restrict__ keepWords,
                                                       const float4* __restrict__ boxes,
                                                       float* __restrict__ out_det,
                                                       float* __restrict__ out_keep) {
    int i = blockIdx.x * blockDim.x + threadIdx.x;   // sorted position
    if (i >= N_BOX) return;
    unsigned j = order[i];                           // original index
    float k = ((keepWords[i >> 5] >> (i & 31)) & 1u) ? 1.f : 0.f;
    float4 b = boxes[j];
    out_keep[j] = k;
    out_det[(size_t)j * 4 + 0] = b.x * k;
    out_det[(size_t)j * 4 + 1] = b.y * k;
    out_det[(size_t)j * 4 + 2] = b.z * k;
    out_det[(size_t)j * 4 + 3] = b.w * k;
}

extern "C" void kernel_launch(void* const* d_in, const int* in_sizes, int n_in,
                              void* d_out, int out_size, void* d_ws, size_t ws_size,
                              hipStream_t stream) {
    const float4* anchors = (const float4*)d_in[0];   // (8192,4)
    const float4* deltas  = (const float4*)d_in[1];   // (8192,4)
    const float*  scores  = (const float*)d_in[2];    // (8192,)

    char* ws = (char*)d_ws;
    float4*   boxes_orig = (float4*)(ws + 0);            // 128 KB
    float4*   bsort      = (float4*)(ws + 131072);       // 128 KB
    float*    areas      = (float*)  (ws + 262144);      //  32 KB
    uint32_t* order      = (uint32_t*)(ws + 294912);     //  32 KB
    uint32_t* keepWords  = (uint32_t*)(ws + 327680);     //   1 KB
    uint32_t* mask       = (uint32_t*)(ws + 328704);     //   8 MB (8192*256 u32)

    float* out_det  = (float*)d_out;                     // (8192,4)
    float* out_keep = (float*)d_out + (size_t)N_BOX * 4; // (8192,) as 0/1 floats

    decode_kernel<<<N_BOX / 256, 256, 0, stream>>>(anchors, deltas, boxes_orig);
    sort_kernel<<<1, 1024, N_BOX * sizeof(unsigned long long), stream>>>(
        scores, boxes_orig, order, bsort, areas);
    dim3 mgrid(N_BOX / 32, N_BOX / 16);   // (256 col-words, 512 row-tiles)
    mask_kernel<<<mgrid, 32, 0, stream>>>(bsort, areas, mask);
    sweep_kernel<<<1, 32, 0, stream>>>(mask, keepWords);
    finalize_kernel<<<N_BOX / 256, 256, 0, stream>>>(order, keepWords, boxes_orig,
                                                     out_det, out_keep);
}